// MMoE_78030965833967
// MI455X (gfx1250) — compile-verified
//
#include <hip/hip_runtime.h>
#include <math.h>

typedef __attribute__((ext_vector_type(2))) float v2f;
typedef __attribute__((ext_vector_type(8))) float v8f;

#define B_N 16384
#define D_N 1024
#define E_N 8
#define T_N 4
#define H_N 512
#define NCOL 64   // cols 0..31: gate logits (t*8+e), cols 32..63: folded expert dots

// ---------------------------------------------------------------------------
// Kernel A: fold expert weights through task heads, and pack the combined
// [D x 64] weight matrix into WMMA-friendly (k-pair, n) interleaved layout:
//   W2p[((d>>1)*64 + n)*2 + (d&1)] = W2[d][n]
// so a lane's B-fragment (two consecutive K values for one column) is one b64.
// ---------------------------------------------------------------------------
__global__ void __launch_bounds__(256) mmoe_prep(
    const float* __restrict__ We,   // [E, D, H]
    const float* __restrict__ be,   // [E, H]
    const float* __restrict__ Wg,   // [T, D, E]
    const float* __restrict__ Wt,   // [T, H, 1]
    float* __restrict__ W2p,        // [D/2][64][2]  (256 KB)
    float* __restrict__ cte) {      // [T, E]
  int idx = blockIdx.x * blockDim.x + threadIdx.x;   // 0 .. E*D-1
  if (idx < E_N * D_N) {
    int e = idx >> 10;          // / D_N
    int d = idx & (D_N - 1);
    const float* wrow = We + ((size_t)e * D_N + d) * H_N;
    float a0 = 0.f, a1 = 0.f, a2 = 0.f, a3 = 0.f;
    for (int h = 0; h < H_N; ++h) {
      float w = wrow[h];
      a0 += w * Wt[0 * H_N + h];
      a1 += w * Wt[1 * H_N + h];
      a2 += w * Wt[2 * H_N + h];
      a3 += w * Wt[3 * H_N + h];
    }
    float acc[4] = {a0, a1, a2, a3};
    int base = (d >> 1) * NCOL;
    int par  = d & 1;
#pragma unroll
    for (int t = 0; t < T_N; ++t) {
      // folded expert column (n = 32 + t*8 + e)
      W2p[(base + 32 + t * E_N + e) * 2 + par] = acc[t];
      // gate column (n = t*8 + e)
      W2p[(base + t * E_N + e) * 2 + par] = Wg[((size_t)t * D_N + d) * E_N + e];
    }
  }
  // c[t,e] = sum_h be[e,h] * Wt[t,h]
  if (blockIdx.x == 0 && threadIdx.x < T_N * E_N) {
    int t = threadIdx.x >> 3, e = threadIdx.x & 7;
    float s = 0.f;
    for (int h = 0; h < H_N; ++h) s += be[e * H_N + h] * Wt[t * H_N + h];
    cte[t * E_N + e] = s;
  }
}

// ---------------------------------------------------------------------------
// Kernel B: fused [16384 x 1024] x [1024 x 64] f32 WMMA GEMM + softmax mix.
// 256 threads = 8 waves per block; each wave owns 16 rows x all 64 cols
// (four 16x16 f32 accumulator tiles). K loop: 256 steps of K=4 via
// V_WMMA_F32_16X16X4_F32 (exact f32 — problem is HBM-bound, not compute).
// ---------------------------------------------------------------------------
__global__ void __launch_bounds__(256) mmoe_main(
    const float* __restrict__ x,    // [B, D]
    const float* __restrict__ W2p,  // packed [D/2][64][2]
    const float* __restrict__ cte,  // [T, E]
    const float* __restrict__ bg,   // [T, E]
    const float* __restrict__ bt,   // [T, 1]
    float* __restrict__ out) {      // [T, B, 1]
  __shared__ float lds[8][16][68];  // stride 68 -> conflict-free column reads

  const int wave = threadIdx.x >> 5;
  const int lane = threadIdx.x & 31;
  const int ml   = lane & 15;   // M (A rows) / N (B cols) index within tile
  const int kh   = lane >> 4;   // K half: lanes 16-31 hold K+2,K+3
  const int row0 = blockIdx.x * 128 + wave * 16;

  // A-frag: x[row0+ml][k0 + 2*kh + {0,1}]  (two consecutive floats -> b64)
  const float* ap = x + (size_t)(row0 + ml) * D_N + kh * 2;
  // B-frag base: kk = k0/2 + kh, n = t*16 + ml  -> W2p[(kk*64 + n)*2]
  const float* bp = W2p + ((size_t)kh * NCOL + ml) * 2;

  v8f acc0 = {}, acc1 = {}, acc2 = {}, acc3 = {};

#pragma unroll 4
  for (int k0 = 0; k0 < D_N; k0 += 4) {
    v2f a = *(const v2f*)(ap + k0);
    const float* bk = bp + (size_t)(k0 >> 1) * (NCOL * 2);
    v2f b0 = *(const v2f*)(bk + 0 * 32);
    v2f b1 = *(const v2f*)(bk + 1 * 32);
    v2f b2 = *(const v2f*)(bk + 2 * 32);
    v2f b3 = *(const v2f*)(bk + 3 * 32);
    acc0 = __builtin_amdgcn_wmma_f32_16x16x4_f32(false, a, false, b0, (short)0, acc0, false, false);
    acc1 = __builtin_amdgcn_wmma_f32_16x16x4_f32(false, a, false, b1, (short)0, acc1, false, false);
    acc2 = __builtin_amdgcn_wmma_f32_16x16x4_f32(false, a, false, b2, (short)0, acc2, false, false);
    acc3 = __builtin_amdgcn_wmma_f32_16x16x4_f32(false, a, false, b3, (short)0, acc3, false, false);
  }

  // C/D layout: VGPR j -> row (j + 8*kh), col (tile*16 + ml)
  const int rbase = kh * 8;
#pragma unroll
  for (int j = 0; j < 8; ++j) {
    lds[wave][rbase + j][0 * 16 + ml] = acc0[j];
    lds[wave][rbase + j][1 * 16 + ml] = acc1[j];
    lds[wave][rbase + j][2 * 16 + ml] = acc2[j];
    lds[wave][rbase + j][3 * 16 + ml] = acc3[j];
  }
  __syncthreads();

  // Each wave: 16 rows x 4 tasks = 64 (m,t) pairs -> 2 per lane.
#pragma unroll
  for (int q = 0; q < 2; ++q) {
    int p = lane + 32 * q;   // 0..63
    int m = p >> 2;          // 0..15
    int t = p & 3;
    float lg[8], sv[8];
    float mx = -3.0e38f;
#pragma unroll
    for (int e = 0; e < 8; ++e) {
      lg[e] = lds[wave][m][t * 8 + e] + bg[t * E_N + e];
      sv[e] = lds[wave][m][32 + t * 8 + e] + cte[t * E_N + e];
      mx = fmaxf(mx, lg[e]);
    }
    float den = 0.f, num = 0.f;
#pragma unroll
    for (int e = 0; e < 8; ++e) {
      float w = __expf(lg[e] - mx);
      den += w;
      num += w * sv[e];
    }
    out[(size_t)t * B_N + (row0 + m)] = num / den + bt[t];
  }
}

extern "C" void kernel_launch(void* const* d_in, const int* in_sizes, int n_in,
                              void* d_out, int out_size, void* d_ws, size_t ws_size,
                              hipStream_t stream) {
  const float* x  = (const float*)d_in[0];
  const float* We = (const float*)d_in[1];
  const float* be = (const float*)d_in[2];
  const float* Wg = (const float*)d_in[3];
  const float* bg = (const float*)d_in[4];
  const float* Wt = (const float*)d_in[5];
  const float* bt = (const float*)d_in[6];
  float* out = (float*)d_out;

  float* W2p = (float*)d_ws;              // D*64 floats = 256 KB
  float* cte = W2p + (size_t)D_N * NCOL;  // 32 floats

  mmoe_prep<<<(E_N * D_N) / 256, 256, 0, stream>>>(We, be, Wg, Wt, W2p, cte);
  mmoe_main<<<B_N / 128, 256, 0, stream>>>(x, W2p, cte, bg, bt, out);
}